// Net_21586505630402
// MI455X (gfx1250) — compile-verified
//
#include <hip/hip_runtime.h>

#define NN 100000
#define D  256

typedef float v2f __attribute__((ext_vector_type(2)));
typedef float v8f __attribute__((ext_vector_type(8)));

#if defined(__gfx1250__) && __has_builtin(__builtin_amdgcn_wmma_f32_16x16x4_f32)
#define HAVE_WMMA_F32 1
#else
#define HAVE_WMMA_F32 0
#endif

// ---------------------------------------------------------------------------
// Row gather: out[i, :] = table[idx[i], :]   (64 threads/row, float4 each)
// ---------------------------------------------------------------------------
__global__ void gather_rows(const float* __restrict__ table,
                            const int* __restrict__ idx,
                            float* __restrict__ out, int nrows)
{
    long long tid = (long long)blockIdx.x * blockDim.x + threadIdx.x;
    int row = (int)(tid >> 6);
    if (row >= nrows) return;
    int c = (int)(tid & 63) * 4;
    const float4 v = *(const float4*)(table + (size_t)idx[row] * D + c);
    *(float4*)(out + (size_t)row * D + c) = v;
}

// ---------------------------------------------------------------------------
// Zero fill (grid-stride)
// ---------------------------------------------------------------------------
__global__ void zero_fill(float* __restrict__ p, long long n)
{
    long long stride = (long long)gridDim.x * blockDim.x;
    for (long long i = (long long)blockIdx.x * blockDim.x + threadIdx.x; i < n; i += stride)
        p[i] = 0.0f;
}

// ---------------------------------------------------------------------------
// Edge scatter-add: agg[dst[e], :] += x[src[e], :]; cnt[dst[e]] += 1
// 64 threads per edge, float4 payload, global_atomic_add_f32
// ---------------------------------------------------------------------------
__global__ void scatter_add(const float* __restrict__ x,
                            const int* __restrict__ src,
                            const int* __restrict__ dst,
                            float* __restrict__ agg,
                            float* __restrict__ cnt, int E)
{
    long long tid = (long long)blockIdx.x * blockDim.x + threadIdx.x;
    int e = (int)(tid >> 6);
    if (e >= E) return;
    int c = (int)(tid & 63) * 4;
    int s = src[e];
    int d = dst[e];
    const float4 v = *(const float4*)(x + (size_t)s * D + c);
    float* p = agg + (size_t)d * D + c;
    atomicAdd(p + 0, v.x);
    atomicAdd(p + 1, v.y);
    atomicAdd(p + 2, v.z);
    atomicAdd(p + 3, v.w);
    if ((tid & 63) == 0) atomicAdd(cnt + d, 1.0f);
}

// ---------------------------------------------------------------------------
// mean[i,:] = agg[i,:] / max(cnt[i], 1)   (in place on agg)
// ---------------------------------------------------------------------------
__global__ void mean_div(float* __restrict__ agg,
                         const float* __restrict__ cnt, int nrows)
{
    long long tid = (long long)blockIdx.x * blockDim.x + threadIdx.x;
    int row = (int)(tid >> 6);
    if (row >= nrows) return;
    int c = (int)(tid & 63) * 4;
    float rcp = 1.0f / fmaxf(cnt[row], 1.0f);
    float4 v = *(float4*)(agg + (size_t)row * D + c);
    v.x *= rcp; v.y *= rcp; v.z *= rcp; v.w *= rcp;
    *(float4*)(agg + (size_t)row * D + c) = v;
}

// ---------------------------------------------------------------------------
// Fused SAGE GEMM:  out = mean @ Wl + bl + xin @ Wr
//   Block: 256 threads = 8 waves; each wave computes a 32x32 output tile
//   (4 x v8f accumulators) with V_WMMA_F32_16X16X4_F32. Each k-step:
//   8 WMMAs : 4 global_b64 A fragments : 4 ds_2addr B fragments (2x reuse
//   of every fragment). B panels (Wl/Wr, D x 32 each) staged in LDS (64 KB).
// ---------------------------------------------------------------------------
__global__ __launch_bounds__(256) void sage_gemm(
    const float* __restrict__ mean,   // [N, D]
    const float* __restrict__ xin,    // [N, D]
    const float* __restrict__ Wl,     // [D, D] row-major (k, n)
    const float* __restrict__ bl,     // [D]
    const float* __restrict__ Wr,     // [D, D]
    float* __restrict__ out,          // [N, D]
    int nrows)
{
    __shared__ float sWl[D * 32];
    __shared__ float sWr[D * 32];

    const int colBase = blockIdx.y * 32;

    // Stage B panels: [k=0..255] x [n=0..31], float4 copies (8 vec4 per row)
    for (int i = threadIdx.x; i < D * 8; i += 256) {
        int k  = i >> 3;
        int n4 = (i & 7) * 4;
        *(float4*)&sWl[k * 32 + n4] =
            *(const float4*)&Wl[(size_t)k * D + colBase + n4];
        *(float4*)&sWr[k * 32 + n4] =
            *(const float4*)&Wr[(size_t)k * D + colBase + n4];
    }
    __syncthreads();

    const int wave = threadIdx.x >> 5;
    const int lane = threadIdx.x & 31;
    const int rowBase = (blockIdx.x * 8 + wave) * 32;
    if (rowBase >= nrows) return;     // uniform per wave -> EXEC stays all-ones

    const int nsub = lane & 15;       // output column within 16-wide sub-tile
    const int hi   = lane >> 4;       // lane half selects K pair (A/B layout)

    // C/D layout: VGPR j, lanes 0-15 -> M=j; lanes 16-31 -> M=j+8; N=lane&15.
    v8f acc00, acc01, acc10, acc11;   // [rowSub][colSub]
    const float bias0 = bl[colBase + nsub];
    const float bias1 = bl[colBase + 16 + nsub];
#pragma unroll
    for (int j = 0; j < 8; ++j) {
        acc00[j] = bias0; acc01[j] = bias1;
        acc10[j] = bias0; acc11[j] = bias1;
    }

#if HAVE_WMMA_F32
    // A layout (16x4 f32): lanes 0-15: {A[M][k0],A[M][k0+1]};
    //                      lanes 16-31: {A[M][k0+2],A[M][k0+3]}, M = lane&15.
    const float* mrow0 = mean + (size_t)(rowBase + nsub) * D;
    const float* mrow1 = mean + (size_t)(rowBase + 16 + nsub) * D;
    const float* xrow0 = xin  + (size_t)(rowBase + nsub) * D;
    const float* xrow1 = xin  + (size_t)(rowBase + 16 + nsub) * D;

    for (int k0 = 0; k0 < D; k0 += 4) {
        const int ka = k0 + hi * 2;
        v2f aM0 = { mrow0[ka], mrow0[ka + 1] };
        v2f aM1 = { mrow1[ka], mrow1[ka + 1] };
        v2f aX0 = { xrow0[ka], xrow0[ka + 1] };
        v2f aX1 = { xrow1[ka], xrow1[ka + 1] };
        // B layout (4x16 f32): mirrored K-pair striping, N = lane&15.
        v2f bL0 = { sWl[ka * 32 + nsub],      sWl[(ka + 1) * 32 + nsub] };
        v2f bL1 = { sWl[ka * 32 + 16 + nsub], sWl[(ka + 1) * 32 + 16 + nsub] };
        v2f bR0 = { sWr[ka * 32 + nsub],      sWr[(ka + 1) * 32 + nsub] };
        v2f bR1 = { sWr[ka * 32 + 16 + nsub], sWr[(ka + 1) * 32 + 16 + nsub] };

        acc00 = __builtin_amdgcn_wmma_f32_16x16x4_f32(
                    false, aM0, false, bL0, (short)0, acc00, false, false);
        acc00 = __builtin_amdgcn_wmma_f32_16x16x4_f32(
                    false, aX0, false, bR0, (short)0, acc00, false, false);
        acc01 = __builtin_amdgcn_wmma_f32_16x16x4_f32(
                    false, aM0, false, bL1, (short)0, acc01, false, false);
        acc01 = __builtin_amdgcn_wmma_f32_16x16x4_f32(
                    false, aX0, false, bR1, (short)0, acc01, false, false);
        acc10 = __builtin_amdgcn_wmma_f32_16x16x4_f32(
                    false, aM1, false, bL0, (short)0, acc10, false, false);
        acc10 = __builtin_amdgcn_wmma_f32_16x16x4_f32(
                    false, aX1, false, bR0, (short)0, acc10, false, false);
        acc11 = __builtin_amdgcn_wmma_f32_16x16x4_f32(
                    false, aM1, false, bL1, (short)0, acc11, false, false);
        acc11 = __builtin_amdgcn_wmma_f32_16x16x4_f32(
                    false, aX1, false, bR1, (short)0, acc11, false, false);
    }
#else
    // Scalar fallback (same math, same output mapping)
#pragma unroll 1
    for (int j = 0; j < 8; ++j) {
        const int m0 = rowBase + j + hi * 8;
        const int m1 = m0 + 16;
        const float* mr0 = mean + (size_t)m0 * D;
        const float* mr1 = mean + (size_t)m1 * D;
        const float* xr0 = xin  + (size_t)m0 * D;
        const float* xr1 = xin  + (size_t)m1 * D;
        float s00 = acc00[j], s01 = acc01[j], s10 = acc10[j], s11 = acc11[j];
        for (int k = 0; k < D; ++k) {
            const float wl0 = sWl[k * 32 + nsub], wl1 = sWl[k * 32 + 16 + nsub];
            const float wr0 = sWr[k * 32 + nsub], wr1 = sWr[k * 32 + 16 + nsub];
            s00 += mr0[k] * wl0 + xr0[k] * wr0;
            s01 += mr0[k] * wl1 + xr0[k] * wr1;
            s10 += mr1[k] * wl0 + xr1[k] * wr0;
            s11 += mr1[k] * wl1 + xr1[k] * wr1;
        }
        acc00[j] = s00; acc01[j] = s01; acc10[j] = s10; acc11[j] = s11;
    }
#endif

#pragma unroll
    for (int j = 0; j < 8; ++j) {
        const int m0 = rowBase + j + hi * 8;
        const int m1 = m0 + 16;
        out[(size_t)m0 * D + colBase + nsub]      = acc00[j];
        out[(size_t)m0 * D + colBase + 16 + nsub] = acc01[j];
        out[(size_t)m1 * D + colBase + nsub]      = acc10[j];
        out[(size_t)m1 * D + colBase + 16 + nsub] = acc11[j];
    }
}

// ---------------------------------------------------------------------------
extern "C" void kernel_launch(void* const* d_in, const int* in_sizes, int n_in,
                              void* d_out, int out_size, void* d_ws, size_t ws_size,
                              hipStream_t stream)
{
    const float* emb = (const float*)d_in[0];
    const float* W1l = (const float*)d_in[1];
    const float* b1l = (const float*)d_in[2];
    const float* W1r = (const float*)d_in[3];
    const float* W2l = (const float*)d_in[4];
    const float* b2l = (const float*)d_in[5];
    const float* W2r = (const float*)d_in[6];
    const int* x_idx = (const int*)d_in[7];
    const int* edge  = (const int*)d_in[8];
    const int* drugN = (const int*)d_in[9];
    const int* seN   = (const int*)d_in[10];

    const int N = NN;
    const int E = in_sizes[8] / 2;
    const int* src = edge;
    const int* dst = edge + E;

    // Workspace layout (floats): x0 | h1 | agg | cnt
    float* ws  = (float*)d_ws;
    float* x0  = ws;
    float* h1  = x0 + (size_t)N * D;
    float* agg = h1 + (size_t)N * D;
    float* cnt = agg + (size_t)N * D;

    // Output layout: drug(1024xD) | se(1024xD) | x(NxD)
    float* outDrug = (float*)d_out;
    float* outSe   = outDrug + 1024 * D;
    float* x2      = outSe   + 1024 * D;

    const int threads = 256;
    const int rowGatherBlocks  = (N * 64 + threads - 1) / threads;
    const int edgeBlocks       = (int)(((long long)E * 64 + threads - 1) / threads);
    const long long zeroCount  = (long long)N * D + N;   // agg + cnt contiguous
    dim3 gemmGrid((N / 32 + 7) / 8, D / 32, 1);          // 32-row tiles, 8 waves/block

    // x0 = emb[x_idx]
    gather_rows<<<rowGatherBlocks, threads, 0, stream>>>(emb, x_idx, x0, N);

    // ---- Layer 1 ----
    zero_fill<<<4096, threads, 0, stream>>>(agg, zeroCount);
    scatter_add<<<edgeBlocks, threads, 0, stream>>>(x0, src, dst, agg, cnt, E);
    mean_div<<<rowGatherBlocks, threads, 0, stream>>>(agg, cnt, N);
    sage_gemm<<<gemmGrid, threads, 0, stream>>>(agg, x0, W1l, b1l, W1r, h1, N);

    // ---- Layer 2 ----
    zero_fill<<<4096, threads, 0, stream>>>(agg, zeroCount);
    scatter_add<<<edgeBlocks, threads, 0, stream>>>(h1, src, dst, agg, cnt, E);
    mean_div<<<rowGatherBlocks, threads, 0, stream>>>(agg, cnt, N);
    sage_gemm<<<gemmGrid, threads, 0, stream>>>(agg, h1, W2l, b2l, W2r, x2, N);

    // ---- Final gathers ----
    const int smallBlocks = (1024 * 64 + threads - 1) / threads;
    gather_rows<<<smallBlocks, threads, 0, stream>>>(x2, drugN, outDrug, 1024);
    gather_rows<<<smallBlocks, threads, 0, stream>>>(x2, seN, outSe, 1024);
}